// HeteroGNN_1838246003032
// MI455X (gfx1250) — compile-verified
//
#include <hip/hip_runtime.h>
#include <hip/hip_bf16.h>

// ---------------------------------------------------------------------------
// HeteroGNN (2-layer bipartite TransformerConv, heads=1, D=H=128, ED=32)
// GEMMs (q,k,v,skip) use V_WMMA_F32_16X16X4_F32 with a swizzled-LDS B layout;
// edge phase is memory-bound gather/softmax/scatter with on-the-fly edge
// embedding.
// ---------------------------------------------------------------------------

#define DHID 128
#define EDIM 32

// LDS layout for W: 32 K-groups x 2 halves; each block holds 128 interleaved
// pairs (W[k][n], W[k+1][n]) = 256 floats, padded to 288 floats so that
// half 0 reads LDS banks 0-31 and half 1 reads banks 32-63 (no conflicts).
#define BLK_STRIDE 288
#define LW_FLOATS (64 * BLK_STRIDE)  // 72 KB

typedef float v2f __attribute__((ext_vector_type(2)));
typedef float v8f __attribute__((ext_vector_type(8)));

// ---------------------------------------------------------------------------
// Dense GEMM: C[N x 128] = X[N x 128] @ W[128 x 128] + bias[128]
// 256 threads = 8 waves; each wave owns a 16-row strip; block covers 128 rows.
// WMMA f32 16x16x4 fragment layout (half = lane>=16):
//   A vgpr r : X[rowBase + lane%16][k0 + 2*half + r]
//   B vgpr r : W[k0 + 2*half + r][n0 + lane%16]
//   C vgpr r : C[rowBase + 8*half + r][n0 + lane%16]
// ---------------------------------------------------------------------------
__global__ __launch_bounds__(256) void gemm128_wmma_kernel(
    float* __restrict__ C, const float* __restrict__ X,
    const float* __restrict__ W, const float* __restrict__ bias, int N)
{
    __shared__ float lw[LW_FLOATS];

    const int tid = threadIdx.x;
    // cooperative swizzled stage of W into LDS
    for (int i = tid * 4; i < DHID * DHID; i += 256 * 4) {
        float4 wv = *(const float4*)(&W[i]);
        const int k = i >> 7;    // row of W
        const int n = i & 127;   // col base (multiple of 4, same row for all 4)
        const int base = (k >> 2) * (2 * BLK_STRIDE)
                       + ((k >> 1) & 1) * BLK_STRIDE + (k & 1);
        lw[base + (n + 0) * 2] = wv.x;
        lw[base + (n + 1) * 2] = wv.y;
        lw[base + (n + 2) * 2] = wv.z;
        lw[base + (n + 3) * 2] = wv.w;
    }
    __syncthreads();

    const int wave = tid >> 5;
    const int lane = tid & 31;
    const int ln   = lane & 15;
    const int half = lane >> 4;
    const int rowBase = blockIdx.x * 128 + wave * 16;

    // Clamp the A row: out-of-range rows read row N-1 (valid memory); the
    // garbage only lands in C rows that are guarded at the store.
    int arow = rowBase + ln;
    if (arow >= N) arow = N - 1;
    const float* aPtr = X + (size_t)arow * DHID + 2 * half;

    v8f acc[8] = {};

    for (int k0 = 0; k0 < DHID; k0 += 4) {
        v2f a = *(const v2f*)(&aPtr[k0]);  // global_load_b64
        const float* bbase =
            &lw[(k0 >> 2) * (2 * BLK_STRIDE) + half * BLK_STRIDE + ln * 2];
#pragma unroll
        for (int t = 0; t < 8; ++t) {
            v2f b = *(const v2f*)(&bbase[t * 32]);  // ds_load_b64, aligned pair
            acc[t] = __builtin_amdgcn_wmma_f32_16x16x4_f32(
                false, a, false, b, (short)0, acc[t], false, false);
        }
    }

#pragma unroll
    for (int t = 0; t < 8; ++t) {
        const int n0 = t * 16;
        const float bv = bias[n0 + ln];
#pragma unroll
        for (int r = 0; r < 8; ++r) {
            const int orow = rowBase + half * 8 + r;
            if (orow < N) C[(size_t)orow * DHID + n0 + ln] = acc[t][r] + bv;
        }
    }
}

// ---------------------------------------------------------------------------
// Init / utility kernels
// ---------------------------------------------------------------------------
__global__ void init_ms_kernel(float* __restrict__ m, float* __restrict__ s, int n)
{
    int i = blockIdx.x * blockDim.x + threadIdx.x;
    if (i < n) { m[i] = -__builtin_inff(); s[i] = 0.0f; }
}

__global__ void zero_kernel(float* __restrict__ p, int n)
{
    int i = blockIdx.x * blockDim.x + threadIdx.x;
    if (i < n) p[i] = 0.0f;
}

__global__ void finalize_kernel(float* __restrict__ out, const float* __restrict__ agg,
                                const float* __restrict__ skip, int n, int relu)
{
    int i = blockIdx.x * blockDim.x + threadIdx.x;
    if (i < n) {
        float v = agg[i] + skip[i];
        out[i] = relu ? fmaxf(v, 0.0f) : v;
    }
}

// float atomic max via integer ordering trick (works for mixed signs, -inf init)
__device__ inline void atomicMaxF32(float* addr, float val)
{
    if (val >= 0.0f)
        atomicMax((int*)addr, __float_as_int(val));
    else
        atomicMin((unsigned int*)addr, (unsigned int)__float_as_int(val));
}

// ---------------------------------------------------------------------------
// Pass A: per edge, alpha = dot(q[dst], k[src] + ee) / sqrt(H); running max.
// One wave per edge; lane handles 4 channels; ee computed on the fly.
// ---------------------------------------------------------------------------
__global__ __launch_bounds__(256) void edge_alpha_kernel(
    float* __restrict__ alpha, float* __restrict__ mmax,
    const float* __restrict__ q, const float* __restrict__ k,
    const float* __restrict__ ea, const float* __restrict__ We,
    const float* __restrict__ be,
    const int* __restrict__ src, const int* __restrict__ dst, int nE)
{
    const int e    = (blockIdx.x * blockDim.x + threadIdx.x) >> 5;
    const int lane = threadIdx.x & 31;
    if (e >= nE) return;

    const int s_ = src[e];
    const int d_ = dst[e];
    const int c0 = lane * 4;

    // edge embedding for channels c0..c0+3
    float4 eev = *(const float4*)(&be[c0]);
    const float* eap = ea + (size_t)e * EDIM;
    float ear[EDIM];
#pragma unroll
    for (int j = 0; j < EDIM; j += 4) {
        float4 t = *(const float4*)(&eap[j]);
        ear[j] = t.x; ear[j + 1] = t.y; ear[j + 2] = t.z; ear[j + 3] = t.w;
    }
#pragma unroll
    for (int j = 0; j < EDIM; ++j) {
        float4 w = *(const float4*)(&We[j * DHID + c0]);
        eev.x += ear[j] * w.x; eev.y += ear[j] * w.y;
        eev.z += ear[j] * w.z; eev.w += ear[j] * w.w;
    }

    float4 kv = *(const float4*)(&k[(size_t)s_ * DHID + c0]);
    float4 qv = *(const float4*)(&q[(size_t)d_ * DHID + c0]);
    float p = qv.x * (kv.x + eev.x) + qv.y * (kv.y + eev.y) +
              qv.z * (kv.z + eev.z) + qv.w * (kv.w + eev.w);

#pragma unroll
    for (int off = 16; off > 0; off >>= 1) p += __shfl_xor(p, off, 32);
    p *= 0.08838834764831845f;  // 128^-0.5

    if (lane == 0) {
        alpha[e] = p;
        atomicMaxF32(&mmax[d_], p);
    }
}

// ---------------------------------------------------------------------------
// Pass B: w = exp(alpha - m[dst]); accumulate denominator.
// ---------------------------------------------------------------------------
__global__ void edge_expsum_kernel(float* __restrict__ alpha, float* __restrict__ ssum,
                                   const float* __restrict__ mmax,
                                   const int* __restrict__ dst, int nE)
{
    int e = blockIdx.x * blockDim.x + threadIdx.x;
    if (e >= nE) return;
    int d_ = dst[e];
    float w = __expf(alpha[e] - mmax[d_]);
    alpha[e] = w;
    atomicAdd(&ssum[d_], w);
}

// ---------------------------------------------------------------------------
// Pass C: agg[dst] += (w/s[dst]) * (v[src] + ee). One wave per edge.
// ---------------------------------------------------------------------------
__global__ __launch_bounds__(256) void edge_agg_kernel(
    float* __restrict__ agg, const float* __restrict__ wgt,
    const float* __restrict__ ssum, const float* __restrict__ v,
    const float* __restrict__ ea, const float* __restrict__ We,
    const float* __restrict__ be,
    const int* __restrict__ src, const int* __restrict__ dst, int nE)
{
    const int e    = (blockIdx.x * blockDim.x + threadIdx.x) >> 5;
    const int lane = threadIdx.x & 31;
    if (e >= nE) return;

    const int s_ = src[e];
    const int d_ = dst[e];
    const float coeff = wgt[e] / ssum[d_];
    const int c0 = lane * 4;

    float4 eev = *(const float4*)(&be[c0]);
    const float* eap = ea + (size_t)e * EDIM;
    float ear[EDIM];
#pragma unroll
    for (int j = 0; j < EDIM; j += 4) {
        float4 t = *(const float4*)(&eap[j]);
        ear[j] = t.x; ear[j + 1] = t.y; ear[j + 2] = t.z; ear[j + 3] = t.w;
    }
#pragma unroll
    for (int j = 0; j < EDIM; ++j) {
        float4 w = *(const float4*)(&We[j * DHID + c0]);
        eev.x += ear[j] * w.x; eev.y += ear[j] * w.y;
        eev.z += ear[j] * w.z; eev.w += ear[j] * w.w;
    }

    float4 vv = *(const float4*)(&v[(size_t)s_ * DHID + c0]);
    float* ap = agg + (size_t)d_ * DHID + c0;
    atomicAdd(&ap[0], coeff * (vv.x + eev.x));
    atomicAdd(&ap[1], coeff * (vv.y + eev.y));
    atomicAdd(&ap[2], coeff * (vv.z + eev.z));
    atomicAdd(&ap[3], coeff * (vv.w + eev.w));
}

// ---------------------------------------------------------------------------
// Host-side orchestration
// ---------------------------------------------------------------------------
namespace {

struct Scratch {
    float *q, *k, *v, *skip, *agg, *alpha, *m, *s;
};

static void run_tconv(const float* xsrc, int Nsrc, const float* xdst, int Ndst,
                      const int* srcIdx, const int* dstIdx, const float* ea, int nE,
                      const float* Wq, const float* bq, const float* Wk, const float* bk,
                      const float* Wv, const float* bv, const float* We, const float* be,
                      const float* Ws, const float* bs,
                      float* out, int relu, const Scratch& ws, hipStream_t stream)
{
    const int gemmBlkDst = (Ndst + 127) / 128;
    const int gemmBlkSrc = (Nsrc + 127) / 128;

    gemm128_wmma_kernel<<<gemmBlkDst, 256, 0, stream>>>(ws.q,    xdst, Wq, bq, Ndst);
    gemm128_wmma_kernel<<<gemmBlkSrc, 256, 0, stream>>>(ws.k,    xsrc, Wk, bk, Nsrc);
    gemm128_wmma_kernel<<<gemmBlkSrc, 256, 0, stream>>>(ws.v,    xsrc, Wv, bv, Nsrc);
    gemm128_wmma_kernel<<<gemmBlkDst, 256, 0, stream>>>(ws.skip, xdst, Ws, bs, Ndst);

    init_ms_kernel<<<(Ndst + 255) / 256, 256, 0, stream>>>(ws.m, ws.s, Ndst);
    zero_kernel<<<(Ndst * DHID + 255) / 256, 256, 0, stream>>>(ws.agg, Ndst * DHID);

    const int edgeWaveBlocks = (nE + 7) / 8;  // 8 waves of 32 per 256-thread block
    edge_alpha_kernel<<<edgeWaveBlocks, 256, 0, stream>>>(
        ws.alpha, ws.m, ws.q, ws.k, ea, We, be, srcIdx, dstIdx, nE);
    edge_expsum_kernel<<<(nE + 255) / 256, 256, 0, stream>>>(
        ws.alpha, ws.s, ws.m, dstIdx, nE);
    edge_agg_kernel<<<edgeWaveBlocks, 256, 0, stream>>>(
        ws.agg, ws.alpha, ws.s, ws.v, ea, We, be, srcIdx, dstIdx, nE);

    finalize_kernel<<<(Ndst * DHID + 255) / 256, 256, 0, stream>>>(
        out, ws.agg, ws.skip, Ndst * DHID, relu);
}

}  // namespace

extern "C" void kernel_launch(void* const* d_in, const int* in_sizes, int n_in,
                              void* d_out, int out_size, void* d_ws, size_t ws_size,
                              hipStream_t stream)
{
    const float* x_user = (const float*)d_in[0];
    const float* x_item = (const float*)d_in[1];
    const float* ea_ui  = (const float*)d_in[2];
    const float* ea_iu  = (const float*)d_in[3];
    const float* Wq     = (const float*)d_in[4];
    const float* bq     = (const float*)d_in[5];
    const float* Wk     = (const float*)d_in[6];
    const float* bk     = (const float*)d_in[7];
    const float* Wv     = (const float*)d_in[8];
    const float* bv     = (const float*)d_in[9];
    const float* We     = (const float*)d_in[10];
    const float* be     = (const float*)d_in[11];
    const float* Ws     = (const float*)d_in[12];
    const float* bs     = (const float*)d_in[13];
    const int*   ei_ui  = (const int*)d_in[14];
    const int*   ei_iu  = (const int*)d_in[15];

    const int NU = in_sizes[0] / DHID;
    const int NI = in_sizes[1] / DHID;
    const int E  = in_sizes[2] / EDIM;
    const int Nmax = (NU > NI) ? NU : NI;

    const int* su = ei_ui;      // user -> item
    const int* du = ei_ui + E;
    const int* si = ei_iu;      // item -> user
    const int* di = ei_iu + E;

    // workspace carve-up
    float* p = (float*)d_ws;
    float* xu_mid = p; p += (size_t)NU * DHID;
    float* xi_mid = p; p += (size_t)NI * DHID;
    Scratch ws;
    ws.q     = p; p += (size_t)Nmax * DHID;
    ws.k     = p; p += (size_t)Nmax * DHID;
    ws.v     = p; p += (size_t)Nmax * DHID;
    ws.skip  = p; p += (size_t)Nmax * DHID;
    ws.agg   = p; p += (size_t)Nmax * DHID;
    ws.alpha = p; p += (size_t)E;
    ws.m     = p; p += (size_t)Nmax;
    ws.s     = p; p += (size_t)Nmax;

    float* out_user = (float*)d_out;
    float* out_item = (float*)d_out + (size_t)NU * DHID;

    const size_t WSTRIDE = (size_t)DHID * DHID;  // per (layer, type) weight block
    const size_t BSTRIDE = (size_t)DHID;
    const size_t ESTRIDE = (size_t)EDIM * DHID;

    for (int l = 0; l < 2; ++l) {
        const float* xu = (l == 0) ? x_user : xu_mid;
        const float* xi = (l == 0) ? x_item : xi_mid;
        float* xi_out = (l == 0) ? xi_mid : out_item;
        float* xu_out = (l == 0) ? xu_mid : out_user;
        const int relu = (l == 0) ? 1 : 0;

        const int t0 = l * 2 + 0;  // user->item conv weights
        const int t1 = l * 2 + 1;  // item->user conv weights

        // x_item update: src = users (su), dst = items (du)
        run_tconv(xu, NU, xi, NI, su, du, ea_ui, E,
                  Wq + t0 * WSTRIDE, bq + t0 * BSTRIDE,
                  Wk + t0 * WSTRIDE, bk + t0 * BSTRIDE,
                  Wv + t0 * WSTRIDE, bv + t0 * BSTRIDE,
                  We + t0 * ESTRIDE, be + t0 * BSTRIDE,
                  Ws + t0 * WSTRIDE, bs + t0 * BSTRIDE,
                  xi_out, relu, ws, stream);

        // x_user update: src = items (si), dst = users (di)
        run_tconv(xi, NI, xu, NU, si, di, ea_iu, E,
                  Wq + t1 * WSTRIDE, bq + t1 * BSTRIDE,
                  Wk + t1 * WSTRIDE, bk + t1 * BSTRIDE,
                  Wv + t1 * WSTRIDE, bv + t1 * BSTRIDE,
                  We + t1 * ESTRIDE, be + t1 * BSTRIDE,
                  Ws + t1 * WSTRIDE, bs + t1 * BSTRIDE,
                  xu_out, relu, ws, stream);
    }
}